// ROIAlign_30262339568080
// MI455X (gfx1250) — compile-verified
//
#include <hip/hip_runtime.h>
#include <hip/hip_bf16.h>

// MI455X (gfx1250) fused ROIAlign(7x7,SR=2)+avgpool.
//
// Separable reformulation: out[n,c] = ry_n^T * F[b,c] * rx_n with dense
// per-ROI weight vectors ry/rx built from the 14 sample rows/cols (validity
// mask factorizes, bilinear kernel is separable). Feature map (64MB) is
// L2-resident (192MB L2); kernel is an L2-gather problem -> lanes map to the
// contiguous W axis for 128B-coalesced loads, per-lane fp32 accumulation,
// and the final 32-lane reduction is done exactly in fp32 on the matrix pipe
// with 8 chained V_WMMA_F32_16X16X4_F32 (all-ones B): out[c] = sum_k acc[c,k].

typedef __attribute__((ext_vector_type(2))) float v2f;
typedef __attribute__((ext_vector_type(8))) float v8f;

#define C_DIM 256
#define HW    128
#define NSAMP 14                 // P * SR
#define INV196 (1.0f / 196.0f)

__device__ __forceinline__ float f16rt(float v) {
    // replicate jnp astype(float16).astype(float32): RNE round through fp16
    return (float)(_Float16)v;
}

__global__ __launch_bounds__(32)
void roi_align_pool_kernel(const float* __restrict__ feat,
                           const float* __restrict__ rois,
                           float* __restrict__ out,
                           float* __restrict__ gt)
{
    __shared__ float s_ry[HW];
    __shared__ float s_rx[HW];
    __shared__ float s_acc[16 * 32];   // [channel][lane] fp32 partials

    const int tid = threadIdx.x;       // 0..31, one wave per block
    const int ct  = blockIdx.x;        // channel tile (16 channels)
    const int n   = blockIdx.y;        // ROI index

    for (int i = tid; i < HW; i += 32) { s_ry[i] = 0.0f; s_rx[i] = 0.0f; }
    __syncthreads();

    // ---- per-ROI scalars (uniform across wave) -------------------------
    const float r0 = rois[n * 6 + 0], r1 = rois[n * 6 + 1];
    const float r2 = rois[n * 6 + 2], r3 = rois[n * 6 + 3];
    const float r4 = rois[n * 6 + 4], r5 = rois[n * 6 + 5];
    const int   b  = (int)r0;
    const float x1 = f16rt((r2 - 0.5f * r4) * 128.0f);
    const float x2 = f16rt((r2 + 0.5f * r4) * 128.0f);
    const float y1 = f16rt((r3 - 0.5f * r5) * 128.0f);
    const float y2 = f16rt((r3 + 0.5f * r5) * 128.0f);
    const float sx = fmaxf(x2 - x1, 1.0f) / 7.0f;   // roi_w / P
    const float sy = fmaxf(y2 - y1, 1.0f) / 7.0f;   // roi_h / P

    // ---- build dense ry/rx (lane 0 scatters), bounds computed by all --
    int ylo = HW, yhi = -1, xlo = HW, xhi = -1;
    for (int s = 0; s < NSAMP; ++s) {
        const float off = (float)(s >> 1) + ((float)(s & 1) + 0.5f) * 0.5f;
        const float ysv = y1 + off * sy;
        if (ysv > -1.0f && ysv < 128.0f) {
            const float yc = fminf(fmaxf(ysv, 0.0f), 127.0f);
            const int   y0 = (int)floorf(yc);
            const float ly = yc - (float)y0;
            const int   y1i = min(y0 + 1, HW - 1);
            if (tid == 0) {
                s_ry[y0]  += (1.0f - ly) * INV196;   // fold 1/196 into ry
                s_ry[y1i] += ly * INV196;
            }
            ylo = min(ylo, y0); yhi = max(yhi, y1i);
        }
        const float xsv = x1 + off * sx;
        if (xsv > -1.0f && xsv < 128.0f) {
            const float xc = fminf(fmaxf(xsv, 0.0f), 127.0f);
            const int   x0 = (int)floorf(xc);
            const float lx = xc - (float)x0;
            const int   x1i = min(x0 + 1, HW - 1);
            if (tid == 0) {
                s_rx[x0]  += (1.0f - lx);
                s_rx[x1i] += lx;
            }
            xlo = min(xlo, x0); xhi = max(xhi, x1i);
        }
    }
    __syncthreads();

    // ---- main contraction: acc[c] += ry[y]*rx[x]*F[b,c0+c,y,x] ---------
    float acc[16];
    #pragma unroll
    for (int c = 0; c < 16; ++c) acc[c] = 0.0f;

    const int c0 = ct * 16;
    const float* fbase = feat + ((size_t)b * C_DIM + (size_t)c0) * (HW * HW);

    for (int xc = xlo; xc <= xhi; xc += 32) {         // W-span <= ~54 -> <=2 chunks
        const int xi = xc + tid;
        float wx = 0.0f;
        int xidx = (xhi >= 0) ? xhi : 0;              // safe in-range address
        if (xi <= xhi) { wx = s_rx[xi]; xidx = xi; }  // lane -> contiguous W
        for (int y = ylo; y <= yhi; ++y) {
            const float wy = s_ry[y];                 // uniform
            if (wy == 0.0f) continue;                 // skip gap rows
            const float w = wy * wx;
            const float* p = fbase + y * HW + xidx;
            __builtin_prefetch(p + HW, 0, 0);         // global_prefetch next row
            #pragma unroll
            for (int c = 0; c < 16; ++c)
                acc[c] = fmaf(w, p[c * (HW * HW)], acc[c]);
        }
    }

    // ---- exact fp32 lane reduction on the matrix pipe ------------------
    // Stage acc[c][lane] in LDS, then out[c] = sum_k acc[c][k] via eight
    // chained V_WMMA_F32_16X16X4_F32 with B = ones (full fp32 precision).
    #pragma unroll
    for (int c = 0; c < 16; ++c) s_acc[c * 32 + tid] = acc[c];
    __syncthreads();

    const int m  = tid & 15;       // A-matrix row (channel) held by this lane
    const int hi = tid >> 4;       // lanes 16..31 hold K=+2,+3 of each chunk
    v8f d = {0.f, 0.f, 0.f, 0.f, 0.f, 0.f, 0.f, 0.f};
    const v2f bones = {1.0f, 1.0f};
    #pragma unroll
    for (int q = 0; q < 8; ++q) {
        const int kb = q * 4 + hi * 2;
        v2f a;
        a.x = s_acc[m * 32 + kb];
        a.y = s_acc[m * 32 + kb + 1];
        d = __builtin_amdgcn_wmma_f32_16x16x4_f32(false, a, false, bones,
                                                  (short)0, d, false, false);
    }

    // D layout: lanes 0/16 hold column N=0 -> M=0..7 / M=8..15 in d[0..7]
    float* orow = out + (size_t)n * C_DIM + c0;
    if (tid == 0) {
        *(float4*)(orow + 0) = make_float4(d[0], d[1], d[2], d[3]);
        *(float4*)(orow + 4) = make_float4(d[4], d[5], d[6], d[7]);
    } else if (tid == 16) {
        *(float4*)(orow + 8)  = make_float4(d[0], d[1], d[2], d[3]);
        *(float4*)(orow + 12) = make_float4(d[4], d[5], d[6], d[7]);
    }
    if (ct == 0 && tid == 1) gt[n] = r1;   // second output: rois[:,1]
}

extern "C" void kernel_launch(void* const* d_in, const int* in_sizes, int n_in,
                              void* d_out, int out_size, void* d_ws, size_t ws_size,
                              hipStream_t stream) {
    const float* feat = (const float*)d_in[0];   // (4,256,128,128) fp32
    const float* rois = (const float*)d_in[1];   // (N,6) fp32
    const int N = in_sizes[1] / 6;               // 512
    float* out = (float*)d_out;                  // roi_feats (N,256) then gt (N)
    float* gt  = out + (size_t)N * C_DIM;

    dim3 grid(C_DIM / 16, N);                    // one wave per (ROI, 16-ch tile)
    roi_align_pool_kernel<<<grid, 32, 0, stream>>>(feat, rois, out, gt);
}